// InplaceUpsampler_51453708206621
// MI455X (gfx1250) — compile-verified
//
#include <hip/hip_runtime.h>
#include <hip/hip_bf16.h>

typedef __attribute__((ext_vector_type(2))) float v2f;
typedef __attribute__((ext_vector_type(8))) float v8f;

#define XS_W 144   // input tile stride (% 64 == 16 -> conflict-free B reads)
#define XS_H 20    // 19 halo rows + 1 zero-weight K-pad row (kills the exec-masked tail load)
#define VS_W 148   // vertical-result stride (% 64 == 20 -> conflict-free A reads)
#define VS_H 16

// 1-D tap f = [1,3,3,1]/8, zero outside [0,3]
__device__ __forceinline__ float ftap(int d) {
    return (d == 0 || d == 3) ? 0.125f : ((d == 1 || d == 2) ? 0.375f : 0.0f);
}

// One block: batch n, channel c, 16 output rows x 129 output cols.
// Stage 1: fused pixel-shuffle gather of the halo into LDS (zero pad).
// Stage 2: vertical pass as banded f32 WMMA matmuls  V = Wv(16x20) * X(20x16).
// Stage 3: horizontal pass as banded f32 WMMA matmuls Y = V(16x20) * Wh(20x16).
__global__ __launch_bounds__(256) void InplaceUpsampler_kernel(
        const float* __restrict__ x, float* __restrict__ y) {
    __shared__ float xs[XS_H * XS_W];
    __shared__ float vs[VS_H * VS_W];

    const int t    = blockIdx.x;          // row tile 0..8  (16 rows each, 129 valid)
    const int cb   = blockIdx.y;          // output channel 0..63
    const int n    = blockIdx.z;          // batch 0..31
    const int tid  = threadIdx.x;
    const int lane = tid & 31;
    const int wv   = tid >> 5;            // wave 0..7
    const int hi   = lane >> 4;           // half-wave (K-pair select in WMMA layouts)
    const int ln   = lane & 15;           // M / N index within WMMA tile
    const int oh0  = t * 16;

    // ---------------- stage 1: pixel-shuffle gather -> LDS (zero-padded halo)
    // x_ps[n][cb][H][W] = x[n][cb*4 + (H&1)*2 + (W&1)][H>>1][W>>1]
    const int base_n = (n * 256 + cb * 4) * 4096;
    for (int idx = tid; idx < XS_H * XS_W; idx += 256) {
        const int r = idx / XS_W;
        const int c = idx - r * XS_W;
        const int H = oh0 - 2 + r;        // x_ps row
        const int W = c - 2;              // x_ps col
        float v = 0.0f;
        if (H >= 0 && H < 128 && W >= 0 && W < 128) {
            const int q = ((H & 1) << 1) | (W & 1);
            v = x[base_n + (q << 12) + ((H >> 1) << 6) + (W >> 1)];
        }
        xs[idx] = v;
    }
    // zero the 4 tail columns of vs (cols 144..147, read only by discarded outputs)
    if (tid < 64)
        vs[(tid >> 2) * VS_W + 144 + (tid & 3)] = 0.0f;

    // Band-matrix taps, identical for A-layout (pass 2) and B-layout (pass 3):
    // Wv[m][k] = Wh[k][n] = f[k - lane]; K chunk kc covers k = 4kc + 2*hi + {0,1}.
    v2f wband[5];
    #pragma unroll
    for (int kc = 0; kc < 5; ++kc) {
        const int k0 = kc * 4 + hi * 2;
        wband[kc].x = ftap(k0 - ln);
        wband[kc].y = ftap(k0 + 1 - ln);
    }

    __syncthreads();

    // ---------------- stage 2: vertical banded matmul per 16-col group
    for (int g = wv; g < 9; g += 8) {
        v8f acc = {};
        #pragma unroll
        for (int kc = 0; kc < 5; ++kc) {
            const int k0 = kc * 4 + hi * 2;
            v2f b;   // B: X[k][n=ln], rows k0, k0+1 (row 19 pairs with zero weight)
            b.x = xs[k0 * XS_W + g * 16 + ln];
            b.y = xs[(k0 + 1) * XS_W + g * 16 + ln];
            acc = __builtin_amdgcn_wmma_f32_16x16x4_f32(
                false, wband[kc], false, b, (short)0, acc, false, false);
        }
        // D layout: VGPR v -> row v (lanes 0-15) / row v+8 (lanes 16-31), col = ln
        #pragma unroll
        for (int v = 0; v < 8; ++v)
            vs[(v + hi * 8) * VS_W + g * 16 + ln] = acc[v];
    }

    __syncthreads();

    // ---------------- stage 3: horizontal banded matmul + global store
    const int obase = (n * 64 + cb) * 129 * 129;
    for (int g = wv; g < 9; g += 8) {
        const int c0 = g * 16;
        v8f acc = {};
        #pragma unroll
        for (int kc = 0; kc < 5; ++kc) {
            const int k0 = kc * 4 + hi * 2;
            v2f a;  // A: V[m=ln][kk], kk = k0, k0+1  (vs col index = V col + 2 = c0 + kk)
            a.x = vs[ln * VS_W + c0 + k0];
            a.y = vs[ln * VS_W + c0 + k0 + 1];
            acc = __builtin_amdgcn_wmma_f32_16x16x4_f32(
                false, a, false, wband[kc], (short)0, acc, false, false);
        }
        const int col = c0 + ln;
        #pragma unroll
        for (int v = 0; v < 8; ++v) {
            const int row = oh0 + v + hi * 8;
            if (row < 129 && col < 129)
                y[obase + row * 129 + col] = acc[v];
        }
    }
}

extern "C" void kernel_launch(void* const* d_in, const int* in_sizes, int n_in,
                              void* d_out, int out_size, void* d_ws, size_t ws_size,
                              hipStream_t stream) {
    (void)in_sizes; (void)n_in; (void)out_size; (void)d_ws; (void)ws_size;
    const float* x = (const float*)d_in[0];
    float* y = (float*)d_out;
    dim3 grid(9, 64, 32);   // (row tiles, channels, batch)
    InplaceUpsampler_kernel<<<grid, dim3(256, 1, 1), 0, stream>>>(x, y);
}